// NodeBlock_62801011802180
// MI455X (gfx1250) — compile-verified
//
#include <hip/hip_runtime.h>

#define N_NODES 100000
#define N_EDGES 1600000
#define LATENT 128
#define HIDDEN 128
#define EPS 1e-5f

typedef float v2f __attribute__((ext_vector_type(2)));
typedef float v8f __attribute__((ext_vector_type(8)));

// D = A(16x4 f32) * B(4x16 f32) + C(16x16 f32), full fp32 precision.
__device__ __forceinline__ v8f wmma_f32(v2f a, v2f b, v8f c) {
    return __builtin_amdgcn_wmma_f32_16x16x4_f32(
        /*neg_a=*/false, a, /*neg_b=*/false, b,
        /*c_mod=*/(short)0, c, /*reuse_a=*/false, /*reuse_b=*/false);
}

__global__ void zero_f4_kernel(float4* __restrict__ p, int n4) {
    int i = blockIdx.x * blockDim.x + threadIdx.x;
    if (i < n4) p[i] = make_float4(0.f, 0.f, 0.f, 0.f);
}

// One thread per 4 consecutive floats of one edge row (LATENT/4 = 32 quads per edge).
__global__ void scatter_add_kernel(const float* __restrict__ edge_attr,
                                   const int* __restrict__ receivers,
                                   float* __restrict__ recv_agg) {
    long long i = (long long)blockIdx.x * blockDim.x + threadIdx.x;
    int e = (int)(i >> 5);
    int q = (int)(i & 31);
    if (e >= N_EDGES) return;
    int r = receivers[e];
    float4 v = ((const float4*)(edge_attr + (size_t)e * LATENT))[q];
    float* dst = recv_agg + (size_t)r * LATENT + q * 4;
    atomicAdd(dst + 0, v.x);
    atomicAdd(dst + 1, v.y);
    atomicAdd(dst + 2, v.z);
    atomicAdd(dst + 3, v.w);
}

// 256 threads (8 waves) per block; each block handles 16 nodes.
// Wave w computes output features [16w, 16w+16) for all 16 nodes of the tile.
__global__ __launch_bounds__(256) void mlp_ln_kernel(
    const float* __restrict__ x, const float* __restrict__ recv_agg,
    const float* __restrict__ Ws, const float* __restrict__ bs,
    const float* __restrict__ Wh, const float* __restrict__ bh,
    const float* __restrict__ We, const float* __restrict__ be,
    float* __restrict__ out) {
    __shared__ float hbuf[16][HIDDEN + 4];   // activation tile, padded vs bank conflicts
    __shared__ float mu_s[16], rs_s[16];

    const int tid  = threadIdx.x;
    const int lane = tid & 31;
    const int wave = tid >> 5;          // 0..7
    const int m    = lane & 15;         // tile row for A / output column index within wave slice
    const int hi   = lane >> 4;         // 0: K+0/K+1 half, 1: K+2/K+3 half (f32 A/B layout)
    const int koff = hi * 2;
    const int n    = wave * 16 + m;     // output feature column 0..127
    const size_t node0 = (size_t)blockIdx.x * 16;
    const size_t arow  = node0 + m;     // node row this lane loads for the A fragment

    // ---------- GEMM1: concat(x, recv_agg)[16 x 256] @ Ws[256 x 128] ----------
    v8f acc = {};
    for (int k = 0; k < 2 * LATENT; k += 4) {
        const float* Ap = (k < LATENT) ? (x + arow * LATENT + k)
                                       : (recv_agg + arow * LATENT + (k - LATENT));
        v2f a; a.x = Ap[koff]; a.y = Ap[koff + 1];
        v2f b;
        b.x = Ws[(size_t)(k + koff) * HIDDEN + n];
        b.y = Ws[(size_t)(k + koff + 1) * HIDDEN + n];
        acc = wmma_f32(a, b, acc);
    }
    {
        float bias = bs[n];
        #pragma unroll
        for (int r = 0; r < 8; ++r)     // C layout: VGPR r -> row r (lanes 0-15) / r+8 (16-31)
            hbuf[r + hi * 8][n] = fmaxf(acc[r] + bias, 0.0f);
    }
    __syncthreads();

    // ---------- GEMM2: h[16 x 128] @ Wh[128 x 128] ----------
    v8f acc2 = {};
    for (int k = 0; k < HIDDEN; k += 4) {
        v2f a; a.x = hbuf[m][k + koff]; a.y = hbuf[m][k + koff + 1];
        v2f b;
        b.x = Wh[(size_t)(k + koff) * HIDDEN + n];
        b.y = Wh[(size_t)(k + koff + 1) * HIDDEN + n];
        acc2 = wmma_f32(a, b, acc2);
    }
    __syncthreads();                    // all reads of hbuf done before overwrite
    {
        float bias = bh[n];
        #pragma unroll
        for (int r = 0; r < 8; ++r)
            hbuf[r + hi * 8][n] = fmaxf(acc2[r] + bias, 0.0f);
    }
    __syncthreads();

    // ---------- GEMM3: h[16 x 128] @ We[128 x 128] (no relu) ----------
    v8f acc3 = {};
    for (int k = 0; k < HIDDEN; k += 4) {
        v2f a; a.x = hbuf[m][k + koff]; a.y = hbuf[m][k + koff + 1];
        v2f b;
        b.x = We[(size_t)(k + koff) * HIDDEN + n];
        b.y = We[(size_t)(k + koff + 1) * HIDDEN + n];
        acc3 = wmma_f32(a, b, acc3);
    }
    __syncthreads();
    {
        float bias = be[n];
        #pragma unroll
        for (int r = 0; r < 8; ++r)
            hbuf[r + hi * 8][n] = acc3[r] + bias;
    }
    __syncthreads();

    // ---------- Parameter-free LayerNorm over feature dim ----------
    if (tid < 16) {
        float s = 0.f, ss = 0.f;
        for (int c = 0; c < LATENT; ++c) {
            float v = hbuf[tid][c];
            s += v; ss += v * v;
        }
        float mu  = s * (1.0f / LATENT);
        float var = ss * (1.0f / LATENT) - mu * mu;
        mu_s[tid] = mu;
        rs_s[tid] = rsqrtf(var + EPS);
    }
    __syncthreads();
    {
        int r  = tid >> 4;              // 0..15 node row
        int c0 = (tid & 15) * 8;        // 8-float chunk -> coalesced stores
        float mu = mu_s[r], rs = rs_s[r];
        float* dst = out + (node0 + r) * LATENT + c0;
        #pragma unroll
        for (int j = 0; j < 8; ++j)
            dst[j] = (hbuf[r][c0 + j] - mu) * rs;
    }
}

extern "C" void kernel_launch(void* const* d_in, const int* in_sizes, int n_in,
                              void* d_out, int out_size, void* d_ws, size_t ws_size,
                              hipStream_t stream) {
    const float* x         = (const float*)d_in[0];
    const float* edge_attr = (const float*)d_in[1];
    const int*   receivers = (const int*)d_in[2];
    const int*   senders   = (const int*)d_in[3];
    const float* Ws        = (const float*)d_in[4];
    const float* bs        = (const float*)d_in[5];
    const float* Wh        = (const float*)d_in[6];
    const float* bh        = (const float*)d_in[7];
    const float* We        = (const float*)d_in[8];
    const float* be        = (const float*)d_in[9];

    float* out      = (float*)d_out;
    float* recv_agg = (float*)d_ws;     // N_NODES * LATENT floats = 51.2 MB scratch

    // 1) zero the aggregation buffer
    int n4 = N_NODES * LATENT / 4;
    zero_f4_kernel<<<(n4 + 255) / 256, 256, 0, stream>>>((float4*)recv_agg, n4);

    // 2) scatter-add edge features to receiver nodes
    long long t = (long long)N_EDGES * (LATENT / 4);
    scatter_add_kernel<<<(int)((t + 255) / 256), 256, 0, stream>>>(edge_attr, receivers, recv_agg);

    // 3) fused MLP + LayerNorm (WMMA f32)
    mlp_ln_kernel<<<N_NODES / 16, 256, 0, stream>>>(x, recv_agg, Ws, bs, Wh, bh, We, be, out);

    // 4) pass-through tuple members (raw-bit copies, graph-capture safe)
    float* out_edge = out + (size_t)N_NODES * LATENT;
    hipMemcpyAsync(out_edge, edge_attr, (size_t)N_EDGES * LATENT * sizeof(float),
                   hipMemcpyDeviceToDevice, stream);
    float* out_recv = out_edge + (size_t)N_EDGES * LATENT;
    hipMemcpyAsync(out_recv, receivers, (size_t)N_EDGES * sizeof(int),
                   hipMemcpyDeviceToDevice, stream);
    float* out_send = out_recv + N_EDGES;
    hipMemcpyAsync(out_send, senders, (size_t)N_EDGES * sizeof(int),
                   hipMemcpyDeviceToDevice, stream);
}